// CTCLayer_47382079209959
// MI455X (gfx1250) — compile-verified
//
#include <hip/hip_runtime.h>

// CTC forward (negative log-likelihood), B=128, T=1024, V=512, L=128, S=257.
// One wave32 per batch row; alpha in registers (9 states/lane, shfl_up for
// neighbors). y_pred rows staged DEPTH ahead via gfx1250 async global->LDS
// copies (coalesced b128), throttled with s_wait_asynccnt. LDS gather + log
// are software-pipelined one step ahead so the serial T-scan's critical path
// is only the flattened 3-way logaddexp recurrence.

#define NEGV (-1.0e30f)
#define EPSV (1e-7f)

constexpr int B_ = 128;
constexpr int T_ = 1024;
constexpr int V_ = 512;
constexpr int L_ = 128;
constexpr int S_ = 2 * L_ + 1;   // 257
constexpr int CH = 9;            // alpha states per lane (29 lanes * 9 >= 257)
constexpr int NBUF = 4;          // LDS row ring
constexpr int DEPTH = 3;         // rows in flight (4 async b128 ops each)

typedef __attribute__((address_space(3))) float lds_f;

__device__ __forceinline__ float lae(float x, float y) {
    float mx = fmaxf(x, y);
    float mn = fminf(x, y);
    return mx + __logf(1.0f + __expf(mn - mx));
}

__device__ __forceinline__ void issue_row(const float* grow, float* lrow, int lane) {
    // Copy one 2048B row (512 f32) global->LDS: 4 x b128, 512B coalesced each.
#pragma unroll
    for (int c = 0; c < 4; ++c) {
        const float* gs = grow + lane * 4 + c * 128;
        lds_f* ls = (lds_f*)(lrow + lane * 4 + c * 128);
        asm volatile("global_load_async_to_lds_b128 %0, %1, off"
                     :: "v"(ls), "v"(gs) : "memory");
    }
}

__global__ __launch_bounds__(32) void ctc_fwd_kernel(
    const int* __restrict__ labels,      // [B, L]
    const int* __restrict__ lab_len,     // [B, 1]
    const float* __restrict__ pred,      // [B, T, V] softmax probs
    const int* __restrict__ in_len_p,    // [B, 1]
    float* __restrict__ out)             // [B, 1]
{
    __shared__ float rowbuf[NBUF][V_];   // 8 KB ring of prob rows
    __shared__ float alpha_sh[S_];

    const int b = blockIdx.x;
    const int lane = threadIdx.x;        // wave32
    const int base = lane * CH;
    const int in_len = in_len_p[b];

    // Extended label sequence (blank = V-1 at even s) and skip-add mask.
    // skadd[j] = 0 if the skip transition (s-2) is allowed, else -1e30 so the
    // corresponding exp underflows to exactly 0 in the flattened logaddexp.
    int   extv[CH];
    float skadd[CH];
#pragma unroll
    for (int j = 0; j < CH; ++j) {
        int s = base + j;
        int e = V_ - 1;
        if (s < S_ && (s & 1)) e = labels[b * L_ + ((s - 1) >> 1)];
        extv[j] = e;
        float sk = NEGV;
        if (s < S_ && (s & 1) && s >= 2) {
            int em2 = labels[b * L_ + ((s - 3) >> 1)];   // ext[s-2], s odd >= 3
            if (e != (V_ - 1) && e != em2) sk = 0.0f;
        }
        skadd[j] = sk;
    }

    float a[CH];
#pragma unroll
    for (int j = 0; j < CH; ++j) a[j] = (base + j == 0) ? 0.0f : NEGV;

    const float* prow = pred + (size_t)b * T_ * V_;

    // Prologue: DEPTH rows in flight (DEPTH*4 = 12 async ops).
#pragma unroll
    for (int p = 0; p < DEPTH; ++p)
        issue_row(prow + (size_t)p * V_, rowbuf[p], lane);

    // Row 0 ready once asynccnt <= 4*(DEPTH-1).
    asm volatile("s_wait_asynccnt 8" ::: "memory");
    float lp[CH];
#pragma unroll
    for (int j = 0; j < CH; ++j)
        lp[j] = __logf(rowbuf[0][extv[j]] + EPSV);

#pragma unroll 4
    for (int t = 0; t < T_; ++t) {
        // Keep DEPTH rows in flight; clamped tail writes a buffer never read.
        int tn = t + DEPTH; if (tn > T_ - 1) tn = T_ - 1;
        issue_row(prow + (size_t)tn * V_, rowbuf[(t + DEPTH) % NBUF], lane);

        // Pipeline: gather + log row t+1 while the recurrence below runs on t.
        float lpn[CH];
#pragma unroll
        for (int j = 0; j < CH; ++j) lpn[j] = 0.0f;
        if (t + 1 < T_) {
            // In flight: groups t+1..t+DEPTH -> wait <= 8 completes group t+1.
            asm volatile("s_wait_asynccnt 8" ::: "memory");
            const float* rb = rowbuf[(t + 1) % NBUF];
#pragma unroll
            for (int j = 0; j < CH; ++j)
                lpn[j] = __logf(rb[extv[j]] + EPSV);
        }

        float pl  = __shfl_up(a[CH - 1], 1, 32);   // alpha[base-1]
        float pl2 = __shfl_up(a[CH - 2], 1, 32);   // alpha[base-2]
        if (lane == 0) { pl = NEGV; pl2 = NEGV; }

        const bool upd = (t < in_len);
        float na[CH];
#pragma unroll
        for (int j = 0; j < CH; ++j) {
            float am1 = (j == 0) ? pl  : a[j - 1];
            float am2 = (j == 0) ? pl2 : ((j == 1) ? pl : a[j - 2]);
            float a2e = am2 + skadd[j];            // disallowed skip -> ~-2e30
            float m3  = fmaxf(a[j], fmaxf(am1, a2e));
            float e   = __expf(a[j] - m3) + __expf(am1 - m3) + __expf(a2e - m3);
            float v   = m3 + __logf(e) + lp[j];
            na[j] = upd ? v : a[j];
        }
#pragma unroll
        for (int j = 0; j < CH; ++j) { a[j] = na[j]; lp[j] = lpn[j]; }
    }

    // Publish alpha, lane 0 emits the loss.
#pragma unroll
    for (int j = 0; j < CH; ++j) {
        int s = base + j;
        if (s < S_) alpha_sh[s] = a[j];
    }
    __syncthreads();
    if (lane == 0) {
        int lab = lab_len[b];
        float a_last = alpha_sh[2 * lab];
        float a_prev = alpha_sh[2 * lab - 1];
        out[b] = -lae(a_last, a_prev);
    }
}

extern "C" void kernel_launch(void* const* d_in, const int* in_sizes, int n_in,
                              void* d_out, int out_size, void* d_ws, size_t ws_size,
                              hipStream_t stream) {
    (void)in_sizes; (void)n_in; (void)out_size; (void)d_ws; (void)ws_size;
    const int*   labels  = (const int*)d_in[0];    // true_labels  [B, L] int32
    const int*   lab_len = (const int*)d_in[1];    // true_lengths [B, 1] int32
    const float* pred    = (const float*)d_in[2];  // predicted_labels [B, T, V] f32
    const int*   in_len  = (const int*)d_in[3];    // predicted_lengths [B, 1] int32
    float*       out     = (float*)d_out;          // [B, 1] f32

    hipLaunchKernelGGL(ctc_fwd_kernel, dim3(B_), dim3(32), 0, stream,
                       labels, lab_len, pred, in_len, out);
}